// GNN_maker_16707422781847
// MI455X (gfx1250) — compile-verified
//
#include <hip/hip_runtime.h>
#include <math.h>

#define D 128
#define SCAN_T 1024

typedef __attribute__((ext_vector_type(2))) float v2f;
typedef __attribute__((ext_vector_type(8))) float v8f;

// ---------------------------------------------------------------------------
// GEMM: h[n, j] = sum_k x[n, k] * W[j, k] + b[j]
// One block = 16-row stripe, 8 waves; wave w -> 16-col tile [16w, 16w+16).
// V_WMMA_F32_16X16X4_F32 chained over K=128 (32 WMMAs per tile).
// ---------------------------------------------------------------------------
__global__ __launch_bounds__(256) void gcn_gemm_xWt_bias(
    const float* __restrict__ x, const float* __restrict__ W,
    const float* __restrict__ bias, float* __restrict__ h, int nrows)
{
    const int lane    = threadIdx.x & 31;
    const int wave    = threadIdx.x >> 5;
    const int rowBase = blockIdx.x * 16;
    if (rowBase >= nrows) return;          // block-uniform; EXEC stays all-1s
    const int colBase = wave * 16;

    const int mn   = lane & 15;            // A: row M;  B/C/D: col N
    const int koff = (lane >> 4) << 1;     // lanes 0-15 -> K {k,k+1}; 16-31 -> {k+2,k+3}

    const float* arow = x + (size_t)(rowBase + mn) * D;   // A row (M = mn)
    const float* brow = W + (size_t)(colBase + mn) * D;   // B col n: B[k][n] = W[n][k]
    const float  bn   = bias[colBase + mn];

    v8f c;
#pragma unroll
    for (int r = 0; r < 8; ++r) c[r] = bn; // bias per output column N

#pragma unroll 8
    for (int k = 0; k < D; k += 4) {
        v2f a = *(const v2f*)(arow + k + koff);
        v2f b = *(const v2f*)(brow + k + koff);
        c = __builtin_amdgcn_wmma_f32_16x16x4_f32(
                false, a, false, b, (short)0, c, false, false);
    }

    const int rAdd = (lane >> 4) << 3;     // D: M = r + (lane>=16 ? 8 : 0), N = lane%16
    float* orow = h + (size_t)(rowBase + rAdd) * D + colBase + mn;
#pragma unroll
    for (int r = 0; r < 8; ++r) orow[(size_t)r * D] = c[r];
}

// ---------------------------------------------------------------------------
// CSR build step 1: in-degree histogram (int atomics, once per launch).
// ---------------------------------------------------------------------------
__global__ __launch_bounds__(256) void gcn_degree(
    const int* __restrict__ dst, unsigned* __restrict__ deg, int nEdges)
{
    const int e = blockIdx.x * blockDim.x + threadIdx.x;
    if (e < nEdges) atomicAdd(&deg[dst[e]], 1u);
}

// ---------------------------------------------------------------------------
// CSR build step 2: exclusive prefix sum over nNodes degrees.
// Single workgroup: per-thread serial chunk + Hillis-Steele LDS scan.
// ---------------------------------------------------------------------------
__global__ __launch_bounds__(SCAN_T) void gcn_scan(
    const unsigned* __restrict__ deg, unsigned* __restrict__ rowPtr, int n)
{
    __shared__ unsigned lds[SCAN_T];
    const int t     = threadIdx.x;
    const int chunk = (n + SCAN_T - 1) / SCAN_T;
    const int beg   = t * chunk;
    const int end   = min(n, beg + chunk);

    unsigned mySum = 0;
    for (int i = beg; i < end; ++i) mySum += deg[i];
    lds[t] = mySum;
    __syncthreads();

    for (int off = 1; off < SCAN_T; off <<= 1) {
        unsigned v = (t >= off) ? lds[t - off] : 0u;
        __syncthreads();
        lds[t] += v;
        __syncthreads();
    }

    unsigned base = lds[t] - mySum;        // exclusive prefix for this chunk
    for (int i = beg; i < end; ++i) {
        rowPtr[i] = base;
        base += deg[i];
    }
    if (t == 0) rowPtr[n] = lds[SCAN_T - 1];
}

// ---------------------------------------------------------------------------
// CSR build step 3: bucket fill. eIdx[rowPtr[dst]+pos] = src.
// ---------------------------------------------------------------------------
__global__ __launch_bounds__(256) void gcn_fill(
    const int* __restrict__ src, const int* __restrict__ dst,
    const unsigned* __restrict__ rowPtr, unsigned* __restrict__ cursor,
    int* __restrict__ eIdx, int nEdges)
{
    const int e = blockIdx.x * blockDim.x + threadIdx.x;
    if (e >= nEdges) return;
    const int dn = dst[e];
    const unsigned pos = atomicAdd(&cursor[dn], 1u);
    eIdx[rowPtr[dn] + pos] = src[e];
}

// ---------------------------------------------------------------------------
// Aggregation (gather form, atomic-free): out[n,:] = [tanh] sum_{s in in(n)} h[s,:]
// One wave per destination node; 32 lanes x float4 = 128 features.
// Gathers are coalesced 512B rows from L2-resident h; output written once.
// ---------------------------------------------------------------------------
template <bool DO_TANH>
__global__ __launch_bounds__(256) void gcn_aggregate(
    const float* __restrict__ h, const unsigned* __restrict__ rowPtr,
    const int* __restrict__ eIdx, float* __restrict__ out, int nNodes)
{
    const int wid = (blockIdx.x * blockDim.x + threadIdx.x) >> 5;  // node id
    if (wid >= nNodes) return;             // wave-uniform exit
    const int q = threadIdx.x & 31;

    const unsigned beg = rowPtr[wid];
    const unsigned end = rowPtr[wid + 1];

    float4 acc = make_float4(0.f, 0.f, 0.f, 0.f);
    const float* hq = h + q * 4;
    for (unsigned j = beg; j < end; ++j) {
        const int s = eIdx[j];             // wave-uniform -> scalarized
        const float4 v = *(const float4*)(hq + (size_t)s * D);
        acc.x += v.x; acc.y += v.y; acc.z += v.z; acc.w += v.w;
    }
    if (DO_TANH) {
        acc.x = tanhf(acc.x); acc.y = tanhf(acc.y);
        acc.z = tanhf(acc.z); acc.w = tanhf(acc.w);
    }
    *(float4*)(out + (size_t)wid * D + q * 4) = acc;
}

extern "C" void kernel_launch(void* const* d_in, const int* in_sizes, int n_in,
                              void* d_out, int out_size, void* d_ws, size_t ws_size,
                              hipStream_t stream)
{
    // setup_inputs order: t, y, src, dst, W1, b1, W2, b2, W3, b3
    const float* y  = (const float*)d_in[1];
    const int*  src = (const int*)  d_in[2];
    const int*  dst = (const int*)  d_in[3];
    const float* W1 = (const float*)d_in[4];
    const float* b1 = (const float*)d_in[5];
    const float* W2 = (const float*)d_in[6];
    const float* b2 = (const float*)d_in[7];
    const float* W3 = (const float*)d_in[8];
    const float* b3 = (const float*)d_in[9];

    float* out = (float*)d_out;

    const int nNodes = in_sizes[1] / D;
    const int nEdges = in_sizes[2];

    // ---- workspace carve-up (256B aligned) ----
    char* ws = (char*)d_ws;
    size_t off = 0;
    auto carve = [&](size_t bytes) {
        char* p = ws + off;
        off += (bytes + 255) & ~(size_t)255;
        return p;
    };
    float*    h      = (float*)   carve((size_t)nNodes * D * sizeof(float)); // 25.6 MB
    unsigned* rowPtr = (unsigned*)carve((size_t)(nNodes + 1) * sizeof(unsigned));
    unsigned* deg    = (unsigned*)carve((size_t)nNodes * sizeof(unsigned)); // reused as cursor
    int*      eIdx   = (int*)     carve((size_t)nEdges * sizeof(int));      // 6.4 MB

    const dim3 blk(256);
    const dim3 gemmGrid((nNodes + 15) / 16);
    const dim3 edgeGrid((nEdges + 255) / 256);
    const dim3 aggGrid((unsigned)(((size_t)nNodes * 32 + 255) / 256));

    // ---- build CSR (by destination) once; reused by all 3 layers ----
    hipMemsetAsync(deg, 0, (size_t)nNodes * sizeof(unsigned), stream);
    gcn_degree<<<edgeGrid, blk, 0, stream>>>(dst, deg, nEdges);
    gcn_scan<<<1, SCAN_T, 0, stream>>>(deg, rowPtr, nNodes);
    hipMemsetAsync(deg, 0, (size_t)nNodes * sizeof(unsigned), stream); // cursor = 0
    gcn_fill<<<edgeGrid, blk, 0, stream>>>(src, dst, rowPtr, deg, eIdx, nEdges);

    // ---- layer 1: x1 = tanh(agg(y @ W1^T + b1)) ----
    gcn_gemm_xWt_bias<<<gemmGrid, blk, 0, stream>>>(y, W1, b1, h, nNodes);
    gcn_aggregate<true><<<aggGrid, blk, 0, stream>>>(h, rowPtr, eIdx, out, nNodes);

    // ---- layer 2: x2 = tanh(agg(x1 @ W2^T + b2)) ----
    gcn_gemm_xWt_bias<<<gemmGrid, blk, 0, stream>>>(out, W2, b2, h, nNodes);
    gcn_aggregate<true><<<aggGrid, blk, 0, stream>>>(h, rowPtr, eIdx, out, nNodes);

    // ---- layer 3: out = agg(x2 @ W3^T + b3) (no tanh) ----
    gcn_gemm_xWt_bias<<<gemmGrid, blk, 0, stream>>>(out, W3, b3, h, nNodes);
    gcn_aggregate<false><<<aggGrid, blk, 0, stream>>>(h, rowPtr, eIdx, out, nNodes);
}